// CustomJSD_12352325943644
// MI455X (gfx1250) — compile-verified
//
#include <hip/hip_runtime.h>
#include <hip/hip_bf16.h>
#include <math.h>

// ---------------------------------------------------------------------------
// CustomJSD on MI455X (gfx1250, wave32) -- fully fused single kernel.
//
// One block per row b.  The row's entire working set (2 datasets x T=100
// frames x 32 joints x 3 coords = 76.8KB) is DMA'd into LDS ONCE by the
// Tensor Data Mover (tensor_load_to_lds, TENSORcnt), norms are precomputed
// for all frames, then two barrier-free sweeps run out of LDS:
//   Phase A: WMMA Gram tiles -> squared-distance min/max (block reduce; sqrt
//            is monotonic so it is applied once, after the reduction).
//   Phase B: WMMA Gram tiles -> raw v_sqrt_f32 -> bin -> LDS histograms,
//            then JSD terms + block sum.
// Distances d_ij = sqrt(n_i + n_j - 2*G_ij), G = X X^T (X is 32x3, K padded
// 3->4): each 16x16 tile of G is ONE V_WMMA_F32_16X16X4_F32.  8 wave32 waves
// partition (dataset x quadrant), each sweeping all T frames with no syncs.
// LDS: 76.8K data + 25.6K norms + ~3K reduce/hist = ~105KB of the 320KB WGP.
// ---------------------------------------------------------------------------

typedef __attribute__((ext_vector_type(2))) float     v2f;
typedef __attribute__((ext_vector_type(8))) float     v8f;
typedef __attribute__((ext_vector_type(4))) unsigned  v4u;
typedef __attribute__((ext_vector_type(8))) int       v8i;
typedef __attribute__((ext_vector_type(4))) int       v4i;

#define B_     256
#define T_     100
#define J_     32
#define C_     3
#define NBINS_ 100
#define FRAME_ (J_ * C_)          // 96 floats per frame
#define ROWF_  (T_ * FRAME_)      // 9600 floats per dataset per row
#define NPER_  (T_ * J_ * J_)     // 102400 distances per row per dataset
#define EPS_   1e-8f

// ---------------------------------------------------------------------------
// TDM: 1D row copy (ROWF_ f32 elements) global -> LDS, as a degenerate 2D
// tensor (tile = 1 x ROWF_, data_size = 4B).  D# layout per ISA ch.8:
//  group0: [1:0]=count=1, [63:32]=lds_addr, [120:64]=global_addr,
//          [127:126]=type=2 ("image")
//  group1: [17:16]=data_size=2 (4B), dim0 @ [79:48], dim1 @ [111:80],
//          tile_dim0 @ [127:112], tile_dim1 @ [143:128], stride0 @ [207:160]
// This toolchain exposes the 6-arg builtin:
//   (v4u g0, v8i g1, v4i g2, v4i g3, v8i g4, i32 cpol)
// ---------------------------------------------------------------------------
__device__ __forceinline__ void tdm_load_row(unsigned lds_byte_addr,
                                             const float* gptr) {
  unsigned long long ga = (unsigned long long)(uintptr_t)gptr;
  v4u g0;
  g0[0] = 1u;                                              // count=1, user D#
  g0[1] = lds_byte_addr;                                   // LDS dest (bytes)
  g0[2] = (unsigned)(ga & 0xffffffffu);                    // global_addr lo
  g0[3] = (unsigned)((ga >> 32) & 0x01ffffffu) | (2u << 30); // hi | type=2
  v8i g1;
  g1[0] = (int)(2u << 16);            // wg_mask=0, data_size=2 (4 bytes)
  g1[1] = (int)((unsigned)ROWF_ << 16);  // tensor_dim0[15:0]  (bits 63:48)
  g1[2] = (int)(1u << 16);            // tensor_dim0 hi=0 | tensor_dim1=1
  g1[3] = (int)((unsigned)ROWF_ << 16);  // tensor_dim1 hi=0 | tile_dim0
  g1[4] = 1;                          // tile_dim1=1, tile_dim2=0
  g1[5] = (int)ROWF_;                 // tensor_dim0_stride[31:0]
  g1[6] = 0;                          // stride0 hi | stride1 lo
  g1[7] = 0;                          // stride1 hi
  v4i z4 = (v4i)0;
  v8i z8 = (v8i)0;
  __builtin_amdgcn_tensor_load_to_lds(g0, g1, z4, z4, z8, 0);
}

// One 16x16 Gram tile via WMMA -- BRANCHLESS operand fetch.
// A-matrix 16x4 f32 ISA layout: lanes 0-15 hold (M=lane, K=0..1); lanes
// 16-31 hold (M=lane-16, K=2..3), K=3 zero-padded.  Every lane loads the
// same shape (2 consecutive floats at base + (lane>=16 ? 2 : 0)); the upper
// half-wave's second float is one element past its row (still inside this
// block's LDS allocation) and is discarded by a single v_cndmask -> no
// exec-mask divergence, loads fuse to one 64-bit DS access per operand.
__device__ __forceinline__ v8f gram_tile(const float* __restrict__ X,
                                         int i0, int j0, int lane) {
  const int  sub  = lane & 15;
  const int  koff = (lane >> 4) << 1;     // 0 (lanes 0-15) or 2 (lanes 16-31)
  const bool hi   = lane >= 16;
  const int  abase = (i0 + sub) * C_ + koff;
  const int  bbase = (j0 + sub) * C_ + koff;
  float a0 = X[abase];
  float a1 = X[abase + 1];
  float b0 = X[bbase];
  float b1 = X[bbase + 1];
  v2f a, b;
  a[0] = a0;  a[1] = hi ? 0.0f : a1;      // K=3 zero pad via select
  b[0] = b0;  b[1] = hi ? 0.0f : b1;
  v8f c = {};
  return __builtin_amdgcn_wmma_f32_16x16x4_f32(false, a, false, b,
                                               (short)0, c, false, false);
}

// ---------------------------------------------------------------------------
__global__ __launch_bounds__(256) void jsd_fused(
    const float* __restrict__ g1, const float* __restrict__ g2,
    float* __restrict__ out) {
  __shared__ float    xs[2][ROWF_];       // 76.8KB: whole row, both datasets
  __shared__ float    norms[2][T_ * J_];  // 25.6KB: |x|^2 for every joint
  __shared__ float    red0[256];
  __shared__ float    red1[256];
  __shared__ unsigned hist[2][NBINS_];
  __shared__ float    mnmx[2];

  const int tid  = threadIdx.x;
  const int lane = tid & 31;
  const int wave = tid >> 5;
  const int ds   = wave >> 2;            // 0 = data1, 1 = data2
  const int tile = wave & 3;             // quadrant of the 32x32 matrix
  const int i0   = (tile >> 1) << 4;
  const int j0   = (tile & 1) << 4;
  const int b    = blockIdx.x;

  // ---- Stage whole row via Tensor Data Mover (wave 0 issues + waits) ----
  if (tid < 32) {
    tdm_load_row((unsigned)(uintptr_t)&xs[0][0], g1 + (size_t)b * ROWF_);
    tdm_load_row((unsigned)(uintptr_t)&xs[1][0], g2 + (size_t)b * ROWF_);
    __builtin_amdgcn_s_wait_tensorcnt(0);
  }
  if (tid < NBINS_) {
    hist[0][tid] = 0u;
    hist[1][tid] = 0u;
  }
  __syncthreads();                       // publish LDS row to all waves

  // ---- Precompute all squared norms (2 x T x J = 6400) ------------------
  for (int idx = tid; idx < 2 * T_ * J_; idx += 256) {
    int dd = idx / (T_ * J_);
    int tj = idx - dd * (T_ * J_);
    float x0 = xs[dd][tj * C_ + 0];
    float x1 = xs[dd][tj * C_ + 1];
    float x2 = xs[dd][tj * C_ + 2];
    norms[dd][tj] = x0 * x0 + x1 * x1 + x2 * x2;
  }
  __syncthreads();

  // ---- Phase A: min/max of squared distances (no barriers in loop) ------
  float lmin2 = 3.4e38f;
  float lmax2 = 0.0f;
  for (int t = 0; t < T_; ++t) {
    const float* X = &xs[ds][t * FRAME_];
    const float* N = &norms[ds][t * J_];
    v8f g = gram_tile(X, i0, j0, lane);
    const int   rb = i0 + ((lane >= 16) ? 8 : 0);
    const float nc = N[j0 + (lane & 15)];
#pragma unroll
    for (int r = 0; r < 8; ++r) {
      float d2 = fmaxf(N[rb + r] + nc - 2.0f * g[r], 0.0f);
      lmin2 = fminf(lmin2, d2);
      lmax2 = fmaxf(lmax2, d2);
    }
  }
  red0[tid] = lmin2;
  red1[tid] = lmax2;
  __syncthreads();
#pragma unroll
  for (int s = 128; s > 0; s >>= 1) {
    if (tid < s) {
      red0[tid] = fminf(red0[tid], red0[tid + s]);
      red1[tid] = fmaxf(red1[tid], red1[tid + s]);
    }
    __syncthreads();
  }
  if (tid == 0) {
    mnmx[0] = sqrtf(red0[0]);            // sqrt commutes with min/max
    mnmx[1] = sqrtf(red1[0]);
  }
  __syncthreads();

  const float mn   = mnmx[0];
  const float mx   = mnmx[1];
  const float invw = (float)NBINS_ / fmaxf(mx - mn, 1e-30f);

  // ---- Phase B: histograms (no barriers in loop) ------------------------
  for (int t = 0; t < T_; ++t) {
    const float* X = &xs[ds][t * FRAME_];
    const float* N = &norms[ds][t * J_];
    v8f g = gram_tile(X, i0, j0, lane);
    const int   rb = i0 + ((lane >= 16) ? 8 : 0);
    const float nc = N[j0 + (lane & 15)];
#pragma unroll
    for (int r = 0; r < 8; ++r) {
      float d2 = fmaxf(N[rb + r] + nc - 2.0f * g[r], 0.0f);
      float d  = __builtin_amdgcn_sqrtf(d2);   // raw v_sqrt_f32, TRANS pipe
      int idx = (int)floorf((d - mn) * invw);  // searchsorted(right)-1
      idx = idx < 0 ? 0 : (idx > NBINS_ - 1 ? NBINS_ - 1 : idx);
      atomicAdd(&hist[ds][idx], 1u);
    }
  }
  __syncthreads();

  // ---- JSD over 100 bins ------------------------------------------------
  float term = 0.0f;
  if (tid < NBINS_) {
    const float w     = (mx - mn) * (1.0f / NBINS_);
    const float scale = 1.0f / ((float)NPER_ * w);   // counts.sum() == NPER_
    float px = (float)hist[0][tid] * scale;
    float qx = (float)hist[1][tid] * scale;
    float m  = 0.5f * (px + qx);
    float lm = __logf(m + EPS_);
    term = 0.5f * (px * (__logf(px + EPS_) - lm) +
                   qx * (__logf(qx + EPS_) - lm));
  }
  red0[tid] = term;
  __syncthreads();
#pragma unroll
  for (int s = 128; s > 0; s >>= 1) {
    if (tid < s) red0[tid] += red0[tid + s];
    __syncthreads();
  }
  if (tid == 0) out[b] = red0[0];
}

// ---------------------------------------------------------------------------
extern "C" void kernel_launch(void* const* d_in, const int* in_sizes, int n_in,
                              void* d_out, int out_size, void* d_ws, size_t ws_size,
                              hipStream_t stream) {
  const float* g1 = (const float*)d_in[0];
  const float* g2 = (const float*)d_in[1];
  float* out      = (float*)d_out;
  (void)d_ws; (void)ws_size; (void)in_sizes; (void)n_in; (void)out_size;

  jsd_fused<<<B_, 256, 0, stream>>>(g1, g2, out);
}